// Conv2D_16930761081407
// MI455X (gfx1250) — compile-verified
//
#include <hip/hip_runtime.h>

typedef float v2f __attribute__((ext_vector_type(2)));
typedef float v8f __attribute__((ext_vector_type(8)));

#define XW 4096          // input width/height
#define OW 4081          // output width/height (4096-16+1)
#define ROWS_PER_SEG 128
#define NSTRIPS 256      // ceil(4081/16) column strips of 16
#define NSEGS 32         // ceil(4081/128) row segments

// Swap the two 16-lane halves of a wave32 register in ONE instruction:
// v_permlanex16_b32 with identity selects (lane i <-> lane i^16).
__device__ __forceinline__ float swap_halves(float v) {
  int i = __builtin_bit_cast(int, v);
  int s = __builtin_amdgcn_permlanex16(i, i, 0x76543210u, 0xfedcba98u,
                                       false, false);
  return __builtin_bit_cast(float, s);
}

// One 8-unrolled pipeline step. u is a compile-time literal 0..7 so all ring
// indices fold to constants. DO_STORE selects warm-up vs main-phase body.
#define STEP(u, T, DO_STORE)                                                   \
  {                                                                            \
    const int t = (T) + (u);                                                   \
    const int r = os + t; /* scalar: os is readfirstlane'd */                  \
    float old = acc[(u)&7];                                                    \
    if (DO_STORE) {                                                            \
      const int orow = r - 16;            /* scalar */                         \
      if ((orow < oe) && storelane) {     /* scalar cmp & invariant lanemask */\
        const char* orowb = (const char*)out + (size_t)orow * (OW * 4);        \
        *(float*)(orowb + ocolb) = old + bval;                                 \
      }                                                                        \
    }                                                                          \
    float sw = swap_halves(old);                                               \
    acc[(u)&7] = h ? sw : 0.0f;                                                \
    const char* xrow = (const char*)x + (size_t)min(r, XW - 1) * (XW * 4);     \
    v8f y = {};                                                                \
    _Pragma("unroll") for (int c = 0; c < 4; ++c) {                            \
      v2f b;                                                                   \
      b.x = *(const float*)(xrow + coffb[2 * c]);                              \
      b.y = *(const float*)(xrow + coffb[2 * c + 1]);                          \
      y = __builtin_amdgcn_wmma_f32_16x16x4_f32(false, A[c], false, b,         \
                                                (short)0, y, false, false);    \
    }                                                                          \
    _Pragma("unroll") for (int k = 0; k < 8; ++k) acc[((u)-k) & 7] += y[k];    \
  }

// Each wave owns one 16-column output strip x one row segment.
// Per image row r: Y[p,n] = sum_q w[p,q]*x[r, j0+n+q] computed as
// W(16x16) x Toeplitz(16x16) via 4 chained v_wmma_f32_16x16x4_f32 (K=16),
// then ring-accumulated into 8 full-wave slots: Y VGPR k holds rows p=k
// (lanes0-15) and p=k+8 (lanes16-31), feeding output rows r-k and r-k-8,
// which share slot (t-k)&7. One permlanex16 per step migrates/extracts.
__global__ __launch_bounds__(256) void conv16x16_wmma_f32(
    const float* __restrict__ x, const float* __restrict__ w,
    const float* __restrict__ bias, float* __restrict__ out)
{
  const int tid  = blockIdx.x * blockDim.x + threadIdx.x;
  const int lane = threadIdx.x & 31;
  // Force wave-uniform values into SGPRs so addressing goes SALU + SADDR.
  const int wave  = __builtin_amdgcn_readfirstlane(tid >> 5);
  const int strip = wave & (NSTRIPS - 1);
  const int seg   = wave >> 8;                     // wave / NSTRIPS
  const int j0    = strip * 16;
  const int os    = seg * ROWS_PER_SEG;            // first output row
  const int oe    = min(os + ROWS_PER_SEG, OW);    // past-last output row
  const int n     = lane & 15;
  const int h     = lane >> 4;

  // ---- Preload A = weight (16x16), 4 chunks of K=4.
  // 32-bit A layout: VGPR0 = K0 (lanes0-15) / K2 (lanes16-31), VGPR1 = K1/K3.
  v2f A[4];
#pragma unroll
  for (int c = 0; c < 4; ++c) {
    A[c].x = w[n * 16 + 4 * c + 2 * h];
    A[c].y = w[n * 16 + 4 * c + 2 * h + 1];
  }
  const float bval = bias[0];

  // ---- Per-lane clamped column BYTE offsets (loop-invariant, unsigned ->
  // SADDR + 32-bit VGPR offset form). B per lane = (x[r,t], x[r,t+1]),
  // t = j0 + n + 2h + 4c.
  const int colbase = j0 + n + 2 * h;
  unsigned coffb[8];
#pragma unroll
  for (int c = 0; c < 4; ++c) {
    coffb[2 * c]     = (unsigned)min(colbase + 4 * c,     XW - 1) * 4u;
    coffb[2 * c + 1] = (unsigned)min(colbase + 4 * c + 1, XW - 1) * 4u;
  }

  // ---- Store lane predicate (loop-invariant) + per-lane byte offset.
  const int ocol = j0 + n;
  const bool storelane = (h != 0) && (ocol < OW);
  const unsigned ocolb = (unsigned)ocol * 4u;

  // ---- 8-slot ring accumulator.
  float acc[8];
#pragma unroll
  for (int s = 0; s < 8; ++s) acc[s] = 0.0f;

  // Phase 1: warm-up, t = 0..15 (rows os..os+15), no stores.
#pragma unroll
  for (int u = 0; u < 8; ++u) STEP(u, 0, false);
#pragma unroll
  for (int u = 0; u < 8; ++u) STEP(u, 8, false);

  // Phase 2: main, one completed output row per step (orow = os + t - 16).
  const int steps = ((oe - os) + 7) & ~7;  // padded to unroll factor
  for (int T = 16; T < steps + 16; T += 8) {
#pragma unroll
    for (int u = 0; u < 8; ++u) STEP(u, T, true);
  }
}

extern "C" void kernel_launch(void* const* d_in, const int* in_sizes, int n_in,
                              void* d_out, int out_size, void* d_ws, size_t ws_size,
                              hipStream_t stream) {
  const float* x    = (const float*)d_in[0];
  const float* wgt  = (const float*)d_in[1];
  const float* bias = (const float*)d_in[2];
  float* out = (float*)d_out;
  // NSTRIPS * NSEGS = 8192 waves = 262144 threads = 1024 blocks of 256.
  dim3 grid(1024), block(256);
  hipLaunchKernelGGL(conv16x16_wmma_f32, grid, block, 0, stream,
                     x, wgt, bias, out);
}